// LinearBlendSkinning_13391708029049
// MI455X (gfx1250) — compile-verified
//
#include <hip/hip_runtime.h>
#include <hip/hip_bf16.h>
#include <stdint.h>

// ---------------------------------------------------------------------------
// Sizes (fixed by the reference)
// ---------------------------------------------------------------------------
#define JJ   161
#define PP   1127          // 7*J
#define BB   8
#define VV   131072
#define KK   8
#define LP_W 960           // P - 6 - J

typedef __attribute__((ext_vector_type(2))) float v2f;
typedef __attribute__((ext_vector_type(8))) float v8f;

struct Parents { int p[JJ]; };

// ---------------------------------------------------------------------------
// Kernel 1: params = pose @ transform.T + offsets   via V_WMMA_F32_16X16X4_F32
//   Pose (8x1127, zero-padded col 1127) staged in LDS once -> inner loop is
//   ds_load_b64 + 2x global_load_b32 + 2x cndmask + v_wmma, no EXEC churn.
//   A (16x4): lanes 0-15 -> M=lane, K={k0,k0+1}; lanes 16-31 -> K={k0+2,k0+3}
//   B (4x16): N = lane&15, same K split. D: VGPR r, lanes 0-15 -> row M=r.
// ---------------------------------------------------------------------------
__global__ __launch_bounds__(32)
void gemm_params_wmma(const float* __restrict__ gp, const float* __restrict__ lp,
                      const float* __restrict__ sc, const float* __restrict__ T,
                      const float* __restrict__ toff, float* __restrict__ params) {
  __shared__ __align__(16) float pose[BB][PP + 1];    // 36096 B, col PP == 0
  const int lane = threadIdx.x;

  for (int t = lane; t < BB * 6;    t += 32) pose[t / 6   ][t % 6           ] = gp[t];
  for (int t = lane; t < BB * LP_W; t += 32) pose[t / LP_W][6 + (t % LP_W)  ] = lp[t];
  for (int t = lane; t < BB * JJ;   t += 32) pose[t / JJ  ][6 + LP_W + t % JJ] = sc[t];
  if (lane < BB) pose[lane][PP] = 0.f;
  __syncthreads();

  const int  half    = lane >> 4;
  const int  l16     = lane & 15;
  const int  n       = blockIdx.x * 16 + l16;
  const int  nn      = (n < PP) ? n : (PP - 1);       // clamped: unstored cols may be garbage
  const bool arow_ok = (l16 < BB);
  const float* __restrict__ Arow = pose[arow_ok ? l16 : 0];
  const float* __restrict__ Trow = T + (size_t)nn * PP;

  v8f c = {0.f, 0.f, 0.f, 0.f, 0.f, 0.f, 0.f, 0.f};

#pragma unroll 2
  for (int k0 = 0; k0 + 4 < PP; k0 += 4) {            // k0 = 0..1120, all K valid
    const int ka = k0 + 2 * half;
    const float2 av = *(const float2*)(Arow + ka);    // LDS b64, 8B aligned
    v2f a, bm;
    a[0]  = arow_ok ? av.x : 0.f;
    a[1]  = arow_ok ? av.y : 0.f;
    bm[0] = Trow[ka];
    bm[1] = Trow[ka + 1];
    c = __builtin_amdgcn_wmma_f32_16x16x4_f32(false, a, false, bm,
                                              (short)0, c, false, false);
  }
  {                                                   // tail: K = 1124..1126
    const int ka = 1124 + 2 * half;
    v2f a, bm;
    a[0]  = arow_ok ? Arow[ka] : 0.f;
    a[1]  = arow_ok ? Arow[ka + 1] : 0.f;             // LDS pad supplies 0 at 1127
    bm[0] = Trow[ka];
    bm[1] = (ka + 1 < PP) ? Trow[ka + 1] : 0.f;
    c = __builtin_amdgcn_wmma_f32_16x16x4_f32(false, a, false, bm,
                                              (short)0, c, false, false);
  }

  if (half == 0 && n < PP) {
    const float off = toff[n];
#pragma unroll
    for (int r = 0; r < BB; ++r)                      // rows 0..7 = d[0..7], lanes 0-15
      params[r * PP + n] = c[r] + off;
  }
}

// ---------------------------------------------------------------------------
// Quaternion helpers (match reference formulas exactly)
// ---------------------------------------------------------------------------
__device__ __forceinline__ void qmul(const float* a, const float* b, float* o) {
  o[0] = a[3]*b[0] + a[0]*b[3] + a[1]*b[2] - a[2]*b[1];
  o[1] = a[3]*b[1] - a[0]*b[2] + a[1]*b[3] + a[2]*b[0];
  o[2] = a[3]*b[2] + a[0]*b[1] - a[1]*b[0] + a[2]*b[3];
  o[3] = a[3]*b[3] - a[0]*b[0] - a[1]*b[1] - a[2]*b[2];
}
__device__ __forceinline__ void qrot(const float* q, const float* v, float* o) {
  const float tx = 2.f * (q[1]*v[2] - q[2]*v[1]);
  const float ty = 2.f * (q[2]*v[0] - q[0]*v[2]);
  const float tz = 2.f * (q[0]*v[1] - q[1]*v[0]);
  o[0] = v[0] + q[3]*tx + (q[1]*tz - q[2]*ty);
  o[1] = v[1] + q[3]*ty + (q[2]*tx - q[0]*tz);
  o[2] = v[2] + q[3]*tz + (q[0]*ty - q[1]*tx);
}

// ---------------------------------------------------------------------------
// Kernel 2: sequential skeleton solve (8 batches on 8 lanes, states in LDS)
//           + states_to_matrix -> mat table (8 x 161 x 12 f32) in workspace
// ---------------------------------------------------------------------------
__global__ __launch_bounds__(32)
void skeleton_kernel(const float* __restrict__ params, const float* __restrict__ joff,
                     const float* __restrict__ jrot,  const float* __restrict__ bind,
                     float* __restrict__ mat, Parents par) {
  __shared__ float st[BB][JJ][8];           // 41216 B: (t3, r4, s1) per (b, joint)
  const int b = threadIdx.x;

  if (b < BB) {
    for (int i = 0; i < JJ; ++i) {
      const float* jp = params + b * PP + i * 7;
      float lt[3] = { jp[0] + joff[3*i+0], jp[1] + joff[3*i+1], jp[2] + joff[3*i+2] };
      float sx, cx, sy, cy, sz, cz;
      sincosf(0.5f * jp[3], &sx, &cx);
      sincosf(0.5f * jp[4], &sy, &cy);
      sincosf(0.5f * jp[5], &sz, &cz);
      float e[4] = { sx*cy*cz + cx*sy*sz, cx*sy*cz - sx*cy*sz,
                     cx*cy*sz + sx*sy*cz, cx*cy*cz - sx*sy*sz };
      const float jr[4] = { jrot[4*i+0], jrot[4*i+1], jrot[4*i+2], jrot[4*i+3] };
      float lr[4]; qmul(jr, e, lr);
      const float ls = exp2f(jp[6]);

      float gt[3], gr[4], gs;
      const int p = par.p[i];
      if (p < 0) {
        gt[0]=lt[0]; gt[1]=lt[1]; gt[2]=lt[2];
        gr[0]=lr[0]; gr[1]=lr[1]; gr[2]=lr[2]; gr[3]=lr[3];
        gs = ls;
      } else {
        const float* S = st[b][p];
        const float pr[4] = { S[3], S[4], S[5], S[6] };
        const float ps = S[7];
        qmul(pr, lr, gr);
        float lts[3] = { lt[0]*ps, lt[1]*ps, lt[2]*ps };
        qrot(pr, lts, gt);
        gt[0] += S[0]; gt[1] += S[1]; gt[2] += S[2];
        gs = ps * ls;
      }
      float* D = st[b][i];
      D[0]=gt[0]; D[1]=gt[1]; D[2]=gt[2];
      D[3]=gr[0]; D[4]=gr[1]; D[5]=gr[2]; D[6]=gr[3];
      D[7]=gs;
    }
  }
  __syncthreads();

  for (int t = threadIdx.x; t < BB * JJ; t += 32) {
    const int bb = t / JJ, i = t % JJ;
    const float* bd = bind + i * 8;                  // bind_state (1, J, 8)
    const float br[4] = { -bd[3], -bd[4], -bd[5], bd[6] };
    const float bs = 1.0f / bd[7];
    const float nb[3] = { -bd[0], -bd[1], -bd[2] };
    float bt[3]; qrot(br, nb, bt);
    bt[0]*=bs; bt[1]*=bs; bt[2]*=bs;

    const float* S = st[bb][i];
    const float sq[4] = { S[3], S[4], S[5], S[6] };
    float tr[4]; qmul(sq, br, tr);
    const float ts = S[7] * bs;
    float bts[3] = { bt[0]*S[7], bt[1]*S[7], bt[2]*S[7] };
    float tt[3]; qrot(sq, bts, tt);
    tt[0]+=S[0]; tt[1]+=S[1]; tt[2]+=S[2];

    const float x=tr[0], y=tr[1], z=tr[2], w=tr[3];
    const float twx=2.f*x*w, twy=2.f*y*w, twz=2.f*z*w;
    const float txx=2.f*x*x, txy=2.f*y*x, txz=2.f*z*x;
    const float tyy=2.f*y*y, tyz=2.f*z*y, tzz=2.f*z*z;

    float* m = mat + (size_t)(bb * JJ + i) * 12;     // row-major 3x4
    m[0]=(1.f-(tyy+tzz))*ts; m[1]=(txy-twz)*ts;      m[2] =(txz+twy)*ts;      m[3] =tt[0];
    m[4]=(txy+twz)*ts;       m[5]=(1.f-(txx+tzz))*ts;m[6] =(tyz-twx)*ts;      m[7] =tt[1];
    m[8]=(txz-twy)*ts;       m[9]=(tyz+twx)*ts;      m[10]=(1.f-(txx+tyy))*ts;m[11]=tt[2];
  }
}

// ---------------------------------------------------------------------------
// Kernel 3: skinning (the roofline term, ~22 MB -> ~1 us at 23.3 TB/s).
// Matrix table (61.8 KB) staged into LDS with gfx1250 async global->LDS b128
// ops (ASYNCcnt-tracked, no VGPR round-trip), then b128 LDS gathers.
// ---------------------------------------------------------------------------
__global__ __launch_bounds__(256)
void skin_kernel(const float* __restrict__ mat, const float* __restrict__ wts,
                 const float* __restrict__ verts, const int* __restrict__ sidx,
                 float* __restrict__ out) {
  __shared__ __align__(16) float4 smat[BB * JJ * 3];   // 8*161*12 f32 = 61824 B
  const float4* gm = (const float4*)mat;

  for (int t = threadIdx.x; t < BB * JJ * 3; t += 256) {
    const uint32_t lds_off = (uint32_t)(uintptr_t)(&smat[t]);  // flat LDS addr[31:0] == LDS offset
    const uint64_t gaddr   = (uint64_t)(uintptr_t)(gm + t);
    asm volatile("global_load_async_to_lds_b128 %0, %1, off"
                 :: "v"(lds_off), "v"(gaddr) : "memory");
  }
  asm volatile("s_wait_asynccnt 0x0" ::: "memory");
  __syncthreads();

  const int v = blockIdx.x * 256 + threadIdx.x;
  if (v >= VV) return;

  const int4*   si4 = (const int4*)  (sidx + (size_t)v * KK);  // 32B aligned
  const float4* sw4 = (const float4*)(wts  + (size_t)v * KK);
  const int4   i0 = si4[0], i1 = si4[1];
  const float4 w0 = sw4[0], w1 = sw4[1];
  int   idx[KK] = { i0.x, i0.y, i0.z, i0.w, i1.x, i1.y, i1.z, i1.w };
  float w  [KK] = { w0.x, w0.y, w0.z, w0.w, w1.x, w1.y, w1.z, w1.w };
#pragma unroll
  for (int k = 0; k < KK; ++k)
    idx[k] = (idx[k] < 0) ? 0 : (idx[k] >= JJ ? JJ - 1 : idx[k]);

  const float vx = verts[3*v + 0], vy = verts[3*v + 1], vz = verts[3*v + 2];

#pragma unroll
  for (int b = 0; b < BB; ++b) {
    float a0x=0,a0y=0,a0z=0,a0w=0, a1x=0,a1y=0,a1z=0,a1w=0, a2x=0,a2y=0,a2z=0,a2w=0;
#pragma unroll
    for (int k = 0; k < KK; ++k) {
      const float4* mp = &smat[(b * JJ + idx[k]) * 3];
      const float wk = w[k];
      const float4 r0 = mp[0], r1 = mp[1], r2 = mp[2];  // ds_load_b128 x3
      a0x += wk*r0.x; a0y += wk*r0.y; a0z += wk*r0.z; a0w += wk*r0.w;
      a1x += wk*r1.x; a1y += wk*r1.y; a1z += wk*r1.z; a1w += wk*r1.w;
      a2x += wk*r2.x; a2y += wk*r2.y; a2z += wk*r2.z; a2w += wk*r2.w;
    }
    const size_t o = ((size_t)b * VV + v) * 3;
    out[o + 0] = a0x*vx + a0y*vy + a0z*vz + a0w;
    out[o + 1] = a1x*vx + a1y*vy + a1z*vz + a1w;
    out[o + 2] = a2x*vx + a2y*vy + a2z*vz + a2w;
  }
}

// ---------------------------------------------------------------------------
// Host: reproduce numpy RandomState(1) JOINT_PARENTS (MT19937 + masked
// rejection for int64 randint), pass as by-value kernel arg (graph-safe).
// ---------------------------------------------------------------------------
static void compute_parents(int* out) {
  uint32_t mt[624];
  int mti;
  mt[0] = 1u;
  for (int i = 1; i < 624; ++i)
    mt[i] = 1812433253u * (mt[i-1] ^ (mt[i-1] >> 30)) + (uint32_t)i;
  mti = 624;
  auto gen = [&]() -> uint32_t {
    if (mti >= 624) {
      for (int k = 0; k < 624; ++k) {
        uint32_t y = (mt[k] & 0x80000000u) | (mt[(k+1) % 624] & 0x7fffffffu);
        mt[k] = mt[(k+397) % 624] ^ (y >> 1) ^ ((y & 1u) ? 0x9908b0dfu : 0u);
      }
      mti = 0;
    }
    uint32_t y = mt[mti++];
    y ^= y >> 11;
    y ^= (y << 7)  & 0x9d2c5680u;
    y ^= (y << 15) & 0xefc60000u;
    y ^= y >> 18;
    return y;
  };
  out[0] = -1;
  for (int i = 1; i < JJ; ++i) {
    const uint64_t rng = (uint64_t)(i - 1);      // randint(0, i) -> [0, i-1]
    if (rng == 0ULL) { out[i] = 0; continue; }
    uint64_t mask = rng;
    mask |= mask >> 1;  mask |= mask >> 2;  mask |= mask >> 4;
    mask |= mask >> 8;  mask |= mask >> 16; mask |= mask >> 32;
    uint64_t v;
    do {
      const uint64_t hi = gen();
      const uint64_t lo = gen();
      v = ((hi << 32) | lo) & mask;
    } while (v > rng);
    out[i] = (int)v;
  }
}

extern "C" void kernel_launch(void* const* d_in, const int* in_sizes, int n_in,
                              void* d_out, int out_size, void* d_ws, size_t ws_size,
                              hipStream_t stream) {
  (void)in_sizes; (void)n_in; (void)out_size; (void)ws_size;

  const float* gp    = (const float*)d_in[0];   // global_pose (8,6)
  const float* lp    = (const float*)d_in[1];   // local_pose  (8,960)
  const float* sc    = (const float*)d_in[2];   // scale       (8,161)
  const float* T     = (const float*)d_in[3];   // transform   (1127,1127)
  const float* toff  = (const float*)d_in[4];   // transform_offsets (1127)
  const float* joff  = (const float*)d_in[5];   // joint_offset (161,3)
  const float* jrot  = (const float*)d_in[6];   // joint_rotation (161,4)
  const float* bind  = (const float*)d_in[7];   // bind_state (1,161,8)
  const float* wts   = (const float*)d_in[8];   // skin_weights (V,8)
  const float* verts = (const float*)d_in[9];   // mesh_vertices (V,3)
  const int*   sidx  = (const int*)d_in[10];    // skin_indices (V,8)
  float* out = (float*)d_out;                   // (8, V, 3)

  float* params_ws = (float*)d_ws;              // 8*1127 f32
  float* mat_ws    = params_ws + BB * PP;       // 8*161*12 f32, 16B-aligned

  Parents par;
  compute_parents(par.p);

  gemm_params_wmma<<<(PP + 15) / 16, 32, 0, stream>>>(gp, lp, sc, T, toff, params_ws);
  skeleton_kernel <<<1, 32, 0, stream>>>(params_ws, joff, jrot, bind, mat_ws, par);
  skin_kernel     <<<VV / 256, 256, 0, stream>>>(mat_ws, wts, verts, sidx, out);
}